// compute_MMDLoss_44744969289832
// MI455X (gfx1250) — compile-verified
//
#include <hip/hip_runtime.h>

// MMD loss with RBF kernel, fp32, via V_WMMA_F32_16X16X4_F32 on gfx1250.
// x,y: [20000, 64] fp32. Output: scalar fp32.

typedef float v2f __attribute__((ext_vector_type(2)));
typedef float v8f __attribute__((ext_vector_type(8)));

#define NPTS 20000
#define DIMK 64
#define TB   625              // 32-wide super-tile blocks per dimension (20000/32)
#define TPM  (TB * TB)        // super-tiles per Gram matrix
#define TTOT (3 * TPM)        // total super-tiles (xx, yy, xy)
#define MAIN_BLOCKS 4096
#define WAVES_PER_BLOCK 8

// ---------------------------------------------------------------------------
// Kernel 1: per-row squared norms for x and y.
// ---------------------------------------------------------------------------
__global__ __launch_bounds__(256) void mmd_norms(const float* __restrict__ x,
                                                 const float* __restrict__ y,
                                                 float* __restrict__ nx,
                                                 float* __restrict__ ny) {
    int tid = blockIdx.x * blockDim.x + threadIdx.x;
    if (tid < NPTS) {
        const float* r = x + (size_t)tid * DIMK;
        float s = 0.0f;
#pragma unroll
        for (int k = 0; k < DIMK; ++k) s += r[k] * r[k];
        nx[tid] = s;
    } else if (tid < 2 * NPTS) {
        int t2 = tid - NPTS;
        const float* r = y + (size_t)t2 * DIMK;
        float s = 0.0f;
#pragma unroll
        for (int k = 0; k < DIMK; ++k) s += r[k] * r[k];
        ny[t2] = s;
    }
}

// ---------------------------------------------------------------------------
// Kernel 2: persistent tile kernel. Each wave owns 32x32 output super-tiles
// (2x2 grid of 16x16 WMMA tiles), K-loop over 64 in chunks of 4 using
// v_wmma_f32_16x16x4_f32. Fixup converts dot products into exp(-d2/2),
// applies the per-matrix MMD weight, accumulates per lane; block writes one
// deterministic partial.
// ---------------------------------------------------------------------------
__global__ __launch_bounds__(256) void mmd_tiles(const float* __restrict__ x,
                                                 const float* __restrict__ y,
                                                 const float* __restrict__ nx,
                                                 const float* __restrict__ ny,
                                                 float* __restrict__ partials) {
    const int lane = threadIdx.x & 31;
    const int wid  = threadIdx.x >> 5;
    const int lm   = lane & 15;   // row/col within 16-tile
    const int hi   = lane >> 4;   // half-wave select (K pair / M+8 select)

    const int gw     = blockIdx.x * WAVES_PER_BLOCK + wid;
    const int stride = gridDim.x * WAVES_PER_BLOCK;

    const float c   = 0.72134752044448169f;                 // log2(e)/2
    const float wdiag  =  1.0f / ((float)NPTS * (float)NPTS);
    const float wcross = -2.0f / ((float)NPTS * (float)NPTS);

    float acc_lane = 0.0f;

    for (int tile = gw; tile < TTOT; tile += stride) {
        int mt = tile / TPM;            // 0: xx, 1: yy, 2: xy
        int r  = tile - mt * TPM;
        int ti = r / TB;
        int tj = r - ti * TB;

        const float *P, *Q, *Pn, *Qn;
        float w;
        if (mt == 0)      { P = x; Q = x; Pn = nx; Qn = nx; w = wdiag;  }
        else if (mt == 1) { P = y; Q = y; Pn = ny; Qn = ny; w = wdiag;  }
        else              { P = x; Q = y; Pn = nx; Qn = ny; w = wcross; }

        const int rowBase = ti * 32;
        const int colBase = tj * 32;

        // A-fragment (16x4 f32): lane lm holds row M=lm; VGPR v <-> K = v + 2*hi.
        // B-fragment mirrors with N. Both sides use the same K mapping, so the
        // per-chunk dot products are layout-consistent.
        const float* pa0 = P + (size_t)(rowBase + lm) * DIMK + 2 * hi;
        const float* pa1 = pa0 + 16 * DIMK;
        const float* pb0 = Q + (size_t)(colBase + lm) * DIMK + 2 * hi;
        const float* pb1 = pb0 + 16 * DIMK;

        v8f a00 = {}, a01 = {}, a10 = {}, a11 = {};

#pragma unroll
        for (int k0 = 0; k0 < DIMK; k0 += 4) {
            v2f fa0 = *(const v2f*)(pa0 + k0);
            v2f fa1 = *(const v2f*)(pa1 + k0);
            v2f fb0 = *(const v2f*)(pb0 + k0);
            v2f fb1 = *(const v2f*)(pb1 + k0);
            a00 = __builtin_amdgcn_wmma_f32_16x16x4_f32(false, fa0, false, fb0,
                                                        (short)0, a00, false, false);
            a01 = __builtin_amdgcn_wmma_f32_16x16x4_f32(false, fa0, false, fb1,
                                                        (short)0, a01, false, false);
            a10 = __builtin_amdgcn_wmma_f32_16x16x4_f32(false, fa1, false, fb0,
                                                        (short)0, a10, false, false);
            a11 = __builtin_amdgcn_wmma_f32_16x16x4_f32(false, fa1, false, fb1,
                                                        (short)0, a11, false, false);
        }

        // C/D layout: lane L, VGPR v -> M = v + 8*hi (+16 for tm=1), N = lm (+16 for tn=1).
        float yn0 = Qn[colBase + lm];
        float yn1 = Qn[colBase + 16 + lm];

        float tsum = 0.0f;
#pragma unroll
        for (int v = 0; v < 8; ++v) {
            float xn0 = Pn[rowBase + v + 8 * hi];
            float xn1 = Pn[rowBase + 16 + v + 8 * hi];
            float arg;
            // exp(-max(xn+yn-2t,0)/2) == exp2(min((2t-xn-yn)*log2e/2, 0))
            arg = fminf((2.0f * a00[v] - xn0 - yn0) * c, 0.0f);
            tsum += __builtin_amdgcn_exp2f(arg);
            arg = fminf((2.0f * a01[v] - xn0 - yn1) * c, 0.0f);
            tsum += __builtin_amdgcn_exp2f(arg);
            arg = fminf((2.0f * a10[v] - xn1 - yn0) * c, 0.0f);
            tsum += __builtin_amdgcn_exp2f(arg);
            arg = fminf((2.0f * a11[v] - xn1 - yn1) * c, 0.0f);
            tsum += __builtin_amdgcn_exp2f(arg);
        }
        acc_lane += w * tsum;
    }

    // Deterministic wave reduction (wave32) then cross-wave via LDS.
#pragma unroll
    for (int off = 16; off >= 1; off >>= 1)
        acc_lane += __shfl_xor(acc_lane, off, 32);

    __shared__ float wsum[WAVES_PER_BLOCK];
    if (lane == 0) wsum[wid] = acc_lane;
    __syncthreads();
    if (threadIdx.x == 0) {
        float s = 0.0f;
#pragma unroll
        for (int i = 0; i < WAVES_PER_BLOCK; ++i) s += wsum[i];
        partials[blockIdx.x] = s;
    }
}

// ---------------------------------------------------------------------------
// Kernel 3: single-block deterministic reduction of the per-block partials.
// ---------------------------------------------------------------------------
__global__ __launch_bounds__(256) void mmd_reduce(const float* __restrict__ partials,
                                                  int n, float* __restrict__ out) {
    __shared__ float sm[256];
    float s = 0.0f;
    for (int i = threadIdx.x; i < n; i += 256) s += partials[i];
    sm[threadIdx.x] = s;
    __syncthreads();
    for (int off = 128; off >= 1; off >>= 1) {
        if ((int)threadIdx.x < off) sm[threadIdx.x] += sm[threadIdx.x + off];
        __syncthreads();
    }
    if (threadIdx.x == 0) out[0] = sm[0];
}

extern "C" void kernel_launch(void* const* d_in, const int* in_sizes, int n_in,
                              void* d_out, int out_size, void* d_ws, size_t ws_size,
                              hipStream_t stream) {
    (void)in_sizes; (void)n_in; (void)out_size; (void)ws_size;
    const float* x = (const float*)d_in[0];
    const float* y = (const float*)d_in[1];

    float* ws       = (float*)d_ws;
    float* nx       = ws;                 // 20000 floats
    float* ny       = ws + NPTS;          // 20000 floats
    float* partials = ws + 2 * NPTS;      // MAIN_BLOCKS floats

    mmd_norms<<<(2 * NPTS + 255) / 256, 256, 0, stream>>>(x, y, nx, ny);
    mmd_tiles<<<MAIN_BLOCKS, 256, 0, stream>>>(x, y, nx, ny, partials);
    mmd_reduce<<<1, 256, 0, stream>>>(partials, MAIN_BLOCKS, (float*)d_out);
}